// GNN_68616397521281
// MI455X (gfx1250) — compile-verified
//
#include <hip/hip_runtime.h>

typedef float v2f __attribute__((ext_vector_type(2)));
typedef float v8f __attribute__((ext_vector_type(8)));

#define H 64

// ---------------------------------------------------------------- utilities
__global__ void zero_f32(float* __restrict__ p, long n) {
  long i = (long)blockIdx.x * blockDim.x + threadIdx.x;
  long stride = (long)gridDim.x * blockDim.x;
  for (; i < n; i += stride) p[i] = 0.0f;
}

__global__ void deg_count(const long long* __restrict__ dst, float* __restrict__ deg, int E) {
  int e = blockIdx.x * blockDim.x + threadIdx.x;
  if (e < E) atomicAdd(&deg[dst[e]], 1.0f);
}

__global__ void inv_deg(float* __restrict__ deg, int n) {
  int i = blockIdx.x * blockDim.x + threadIdx.x;
  if (i < n) deg[i] = 1.0f / fmaxf(deg[i], 1.0f);
}

// One wave per edge: lane t handles features 2t, 2t+1 (64 feats / 32 lanes).
__global__ __launch_bounds__(256) void scatter_feat(const float* __restrict__ x,
                                                    const long long* __restrict__ src,
                                                    const long long* __restrict__ dst,
                                                    float* __restrict__ agg, int E) {
  int e = blockIdx.x * 8 + (threadIdx.x >> 5);
  if (e >= E) return;
  int lane = threadIdx.x & 31;
  long long s = src[e];
  long long d = dst[e];
  v2f v = *(const v2f*)(x + (long)s * H + lane * 2);
  float* ap = agg + (long)d * H + lane * 2;
  atomicAdd(ap + 0, v.x);
  atomicAdd(ap + 1, v.y);
}

// ---------------------------------------------------------------- fused SAGE layer
// out[n,64] = relu( (agg[n,:]*invdeg[n]) @ Wl^T + x[n,:] @ Wr^T + b )
// One wave owns a 16-row node tile and all 64 output columns (4 WMMA C-tiles).
// WMMA: V_WMMA_F32_16X16X4_F32, K=64 -> 16 k-steps, 8 WMMAs per step.
__global__ __launch_bounds__(256) void sage_gemm(const float* __restrict__ agg,
                                                 const float* __restrict__ xin,
                                                 const float* __restrict__ invdeg,
                                                 const float* __restrict__ Wl,
                                                 const float* __restrict__ Wr,
                                                 const float* __restrict__ bias,
                                                 float* __restrict__ out, int n) {
  const int lane = threadIdx.x & 31;
  const int wave = threadIdx.x >> 5;
  const int row0 = (blockIdx.x * 8 + wave) * 16;
  if (row0 >= n) return;                 // wave-uniform: EXEC stays all-ones below

  const int r  = lane & 15;              // A: row-in-tile / B: output column-in-tile
  const int kh = lane >> 4;              // K half selector (K = 4*ks + 2*kh + {0,1})
  int row = row0 + r; if (row >= n) row = n - 1;   // clamped tail loads

  const float id = invdeg[row];
  const float* arow = agg + (long)row * H;
  const float* xrow = xin + (long)row * H;
  // B[k][col] = W[col][k]; lane supplies column (16*t + r)
  const float* wl0 = Wl + (long)(r +  0) * H;
  const float* wl1 = Wl + (long)(r + 16) * H;
  const float* wl2 = Wl + (long)(r + 32) * H;
  const float* wl3 = Wl + (long)(r + 48) * H;
  const float* wr0 = Wr + (long)(r +  0) * H;
  const float* wr1 = Wr + (long)(r + 16) * H;
  const float* wr2 = Wr + (long)(r + 32) * H;
  const float* wr3 = Wr + (long)(r + 48) * H;

  v8f acc[4] = {v8f{}, v8f{}, v8f{}, v8f{}};

#pragma unroll
  for (int ks = 0; ks < 16; ++ks) {
    const int ko = ks * 4 + kh * 2;
    v2f ax = *(const v2f*)(xrow + ko);
    v2f am; am.x = arow[ko] * id; am.y = arow[ko + 1] * id;

    v2f b;
    b = *(const v2f*)(wl0 + ko);
    acc[0] = __builtin_amdgcn_wmma_f32_16x16x4_f32(false, am, false, b, (short)0, acc[0], false, false);
    b = *(const v2f*)(wr0 + ko);
    acc[0] = __builtin_amdgcn_wmma_f32_16x16x4_f32(false, ax, false, b, (short)0, acc[0], false, false);
    b = *(const v2f*)(wl1 + ko);
    acc[1] = __builtin_amdgcn_wmma_f32_16x16x4_f32(false, am, false, b, (short)0, acc[1], false, false);
    b = *(const v2f*)(wr1 + ko);
    acc[1] = __builtin_amdgcn_wmma_f32_16x16x4_f32(false, ax, false, b, (short)0, acc[1], false, false);
    b = *(const v2f*)(wl2 + ko);
    acc[2] = __builtin_amdgcn_wmma_f32_16x16x4_f32(false, am, false, b, (short)0, acc[2], false, false);
    b = *(const v2f*)(wr2 + ko);
    acc[2] = __builtin_amdgcn_wmma_f32_16x16x4_f32(false, ax, false, b, (short)0, acc[2], false, false);
    b = *(const v2f*)(wl3 + ko);
    acc[3] = __builtin_amdgcn_wmma_f32_16x16x4_f32(false, am, false, b, (short)0, acc[3], false, false);
    b = *(const v2f*)(wr3 + ko);
    acc[3] = __builtin_amdgcn_wmma_f32_16x16x4_f32(false, ax, false, b, (short)0, acc[3], false, false);
  }

  // D layout: VGPR i -> M = i + 8*kh, N = r (+16*t); add bias, ReLU, store.
#pragma unroll
  for (int t = 0; t < 4; ++t) {
    const int col = t * 16 + r;
    const float bv = bias[col];
#pragma unroll
    for (int i = 0; i < 8; ++i) {
      const int rr = row0 + kh * 8 + i;
      float v = acc[t][i] + bv;
      v = v > 0.0f ? v : 0.0f;
      if (rr < n) out[(long)rr * H + col] = v;
    }
  }
}

// ---------------------------------------------------------------- final 64 -> 10 linear
__global__ __launch_bounds__(256) void final_lin(const float* __restrict__ xin,
                                                 const float* __restrict__ Wlin,
                                                 const float* __restrict__ bias,
                                                 float* __restrict__ out, int n) {
  const int lane = threadIdx.x & 31;
  const int wave = threadIdx.x >> 5;
  const int row0 = (blockIdx.x * 8 + wave) * 16;
  if (row0 >= n) return;

  const int r  = lane & 15;
  const int kh = lane >> 4;
  int row = row0 + r; if (row >= n) row = n - 1;
  const float* xrow = xin + (long)row * H;
  const int wrow = r < 10 ? r : 0;       // clamp: cols 10..15 are never stored
  const float* wp = Wlin + (long)wrow * H;

  v8f acc = v8f{};
#pragma unroll
  for (int ks = 0; ks < 16; ++ks) {
    const int ko = ks * 4 + kh * 2;
    v2f a = *(const v2f*)(xrow + ko);
    v2f b = *(const v2f*)(wp + ko);
    acc = __builtin_amdgcn_wmma_f32_16x16x4_f32(false, a, false, b, (short)0, acc, false, false);
  }

  if (r < 10) {
    const float bv = bias[r];
#pragma unroll
    for (int i = 0; i < 8; ++i) {
      const int rr = row0 + kh * 8 + i;
      if (rr < n) out[(long)rr * 10 + r] = acc[i] + bv;
    }
  }
}

// ---------------------------------------------------------------- driver
extern "C" void kernel_launch(void* const* d_in, const int* in_sizes, int n_in,
                              void* d_out, int out_size, void* d_ws, size_t ws_size,
                              hipStream_t stream) {
  const float*     x  = (const float*)d_in[0];
  const long long* ei = (const long long*)d_in[1];      // int64 [2, E]
  const int N = in_sizes[0] / H;
  const int E = in_sizes[1] / 2;
  const long long* src = ei;
  const long long* dst = ei + E;

  float* ws  = (float*)d_ws;
  float* agg = ws;                         // N*64
  float* h0  = agg + (size_t)N * H;        // N*64
  float* h1  = h0 + (size_t)N * H;         // N*64
  float* dg  = h1 + (size_t)N * H;         // N (deg -> invdeg, in place)

  // degrees (edge-structure only; compute once)
  zero_f32<<<1024, 256, 0, stream>>>(dg, (long)N);
  deg_count<<<(E + 255) / 256, 256, 0, stream>>>(dst, dg, E);
  inv_deg<<<(N + 255) / 256, 256, 0, stream>>>(dg, N);

  const int   widx[4] = {2, 5, 8, 11};
  float*      bufs[4] = {h0, h1, h0, h1};
  const float* cur = x;
  const int gemm_blocks = (N + 127) / 128;

  for (int l = 0; l < 4; ++l) {
    zero_f32<<<4096, 256, 0, stream>>>(agg, (long)N * H);
    scatter_feat<<<(E + 7) / 8, 256, 0, stream>>>(cur, src, dst, agg, E);
    sage_gemm<<<gemm_blocks, 256, 0, stream>>>(agg, cur, dg,
        (const float*)d_in[widx[l]],        // Wl
        (const float*)d_in[widx[l] + 1],    // Wr
        (const float*)d_in[widx[l] + 2],    // b
        bufs[l], N);
    cur = bufs[l];
  }

  final_lin<<<gemm_blocks, 256, 0, stream>>>(cur, (const float*)d_in[14],
                                             (const float*)d_in[15], (float*)d_out, N);
}